// SSD_31396210933788
// MI455X (gfx1250) — compile-verified
//
#include <hip/hip_runtime.h>
#include <hip/hip_bf16.h>

typedef float v2f __attribute__((ext_vector_type(2)));
typedef float v8f __attribute__((ext_vector_type(8)));

#define CONF_THRESH 0.01f
#define NMS_THRESH  0.45f
#define TOP_K       200
#define NCLS        21
#define NFG         20      // foreground classes
#define KPAD        208     // TOP_K padded to multiple of 16
#define NTILE       13      // KPAD / 16
#define MASKW       7       // ceil(KPAD/32) u32 words per row
#define DMAX        8732

// ---------------------------------------------------------------------------
// Kernel 0: zero the output tensor (harness poisons it; graph replays need a
// deterministic rewrite every call).
// ---------------------------------------------------------------------------
__global__ void ssd_zero_kernel(float* __restrict__ out, int n) {
    int g = blockIdx.x * blockDim.x + threadIdx.x;
    for (; g < n; g += gridDim.x * blockDim.x) out[g] = 0.0f;
}

// ---------------------------------------------------------------------------
// Kernel 1: box decode + class softmax.
// boxes : [B, D, 4]  (x1,y1,x2,y2)
// scores: [B, NFG, D]  (class-major for coalesced per-class reads later)
// ---------------------------------------------------------------------------
__global__ void ssd_decode_softmax_kernel(const float* __restrict__ loc,
                                          const float* __restrict__ conf,
                                          const float* __restrict__ dbox,
                                          float* __restrict__ boxes,
                                          float* __restrict__ scores,
                                          int B, int D) {
    int g = blockIdx.x * blockDim.x + threadIdx.x;
    if (g >= B * D) return;
    int b = g / D;
    int d = g - b * D;

    // ---- decode (cx,cy,w,h) -> (x1,y1,x2,y2) ----
    const float* l  = loc  + (size_t)g * 4;
    const float* db = dbox + (size_t)d * 4;
    float cx = db[0] + l[0] * 0.1f * db[2];
    float cy = db[1] + l[1] * 0.1f * db[3];
    float w  = db[2] * expf(l[2] * 0.2f);
    float h  = db[3] * expf(l[3] * 0.2f);
    float* bo = boxes + (size_t)g * 4;
    bo[0] = cx - 0.5f * w;
    bo[1] = cy - 0.5f * h;
    bo[2] = cx + 0.5f * w;
    bo[3] = cy + 0.5f * h;

    // ---- softmax over 21 classes ----
    const float* c = conf + (size_t)g * NCLS;
    float m = c[0];
#pragma unroll
    for (int i = 1; i < NCLS; ++i) m = fmaxf(m, c[i]);
    float e[NCLS];
    float s = 0.0f;
#pragma unroll
    for (int i = 0; i < NCLS; ++i) { e[i] = expf(c[i] - m); s += e[i]; }
    float inv = 1.0f / s;
#pragma unroll
    for (int i = 1; i < NCLS; ++i)
        scores[((size_t)b * NFG + (i - 1)) * D + d] = e[i] * inv;
}

// ---------------------------------------------------------------------------
// Kernel 2: per (batch, class) top-200 + NMS + stable compaction.
// One 256-thread (8-wave) block per (b, cc).
// ---------------------------------------------------------------------------
__global__ void __launch_bounds__(256)
ssd_topk_nms_kernel(const float* __restrict__ boxes,
                    const float* __restrict__ scores,
                    float* __restrict__ out,
                    int B, int D) {
    const int tid  = threadIdx.x;
    const int bc   = blockIdx.x;
    const int b    = bc / NFG;
    const int cc   = bc - b * NFG;        // foreground class 0..19 -> output class cc+1
    const int wave = tid >> 5;
    const int lane = tid & 31;
    const int l15  = lane & 15;

    __shared__ float    s_score[DMAX];
    __shared__ float    s_rv[256];
    __shared__ int      s_ri[256];
    __shared__ float    s_topv[KPAD];
    __shared__ int      s_topi[TOP_K];
    __shared__ float    s_bx1[KPAD], s_by1[KPAD], s_bx2[KPAD], s_by2[KPAD];
    __shared__ float    s_area[KPAD];
    __shared__ unsigned s_mask[KPAD * MASKW];
    __shared__ int      s_pos[TOP_K];

    // ---- phase 1: load scores, apply confidence threshold (-> -1 like ref) ----
    const float* sc = scores + ((size_t)b * NFG + cc) * D;
    for (int d = tid; d < D; d += 256) {
        float v = sc[d];
        s_score[d] = (v > CONF_THRESH) ? v : -1.0f;
    }
    __syncthreads();

    // ---- phase 2: top-200 via iterative parallel argmax (ties -> lowest idx,
    //      matching jax.lax.top_k ordering) ----
    for (int k = 0; k < TOP_K; ++k) {
        float bv = -3e38f;
        int   bi = 0x7FFFFFFF;
        for (int d = tid; d < D; d += 256) {
            float v = s_score[d];
            if (v > bv) { bv = v; bi = d; }  // strict > keeps lowest index
        }
        s_rv[tid] = bv; s_ri[tid] = bi;
        __syncthreads();
        for (int off = 128; off > 0; off >>= 1) {
            if (tid < off) {
                float v2 = s_rv[tid + off];
                int   i2 = s_ri[tid + off];
                if (v2 > s_rv[tid] || (v2 == s_rv[tid] && i2 < s_ri[tid])) {
                    s_rv[tid] = v2; s_ri[tid] = i2;
                }
            }
            __syncthreads();
        }
        if (tid == 0) {
            s_topv[k] = s_rv[0];
            s_topi[k] = s_ri[0];
            if (s_ri[0] >= 0 && s_ri[0] < D) s_score[s_ri[0]] = -3e38f;
        }
        __syncthreads();
    }

    // ---- phase 3: gather top boxes into LDS, pad [200,208) with zeros ----
    if (tid < KPAD) {
        float x1 = 0.f, y1 = 0.f, x2 = 0.f, y2 = 0.f;
        if (tid < TOP_K) {
            int idx = s_topi[tid];
            const float* bo = boxes + ((size_t)b * D + idx) * 4;
            x1 = bo[0]; y1 = bo[1]; x2 = bo[2]; y2 = bo[3];
        } else {
            s_topv[tid] = -3e38f;
        }
        s_bx1[tid] = x1; s_by1[tid] = y1; s_bx2[tid] = x2; s_by2[tid] = y2;
        s_area[tid] = (x2 - x1) * (y2 - y1);
    }
    for (int t = tid; t < KPAD * MASKW; t += 256) s_mask[t] = 0u;
    __syncthreads();

    // ---- phase 4: IoU suppression matrix. Union precursor area[i]+area[j] is
    //      a rank-2 outer product computed on the tensor core:
    //        A(16x4) = [area_i, 1, 0, 0] ; B(4x16) = [[1..1],[area_j..],0,0]
    //      D(16x16) = area_i + area_j  via V_WMMA_F32_16X16X4_F32.
    //      Intersection (unclamped, faithful to ref) on VALU; threshold bits
    //      OR'd into the packed mask via LDS atomics.
    for (int t = wave; t < NTILE * NTILE; t += 8) {
        int ti = t / NTILE;
        int tj = t - ti * NTILE;
        float aA = s_area[ti * 16 + l15];
        float aB = s_area[tj * 16 + l15];
        v2f A, Bm;
        A.x  = (lane < 16) ? aA  : 0.0f;   // K=0 column: area_i
        A.y  = (lane < 16) ? 1.0f : 0.0f;  // K=1 column: ones
        Bm.x = (lane < 16) ? 1.0f : 0.0f;  // K=0 row: ones
        Bm.y = (lane < 16) ? aB  : 0.0f;   // K=1 row: area_j
        v8f Cm = {0.f, 0.f, 0.f, 0.f, 0.f, 0.f, 0.f, 0.f};
#if defined(__has_builtin) && __has_builtin(__builtin_amdgcn_wmma_f32_16x16x4_f32)
        v8f Dm = __builtin_amdgcn_wmma_f32_16x16x4_f32(
            false, A, false, Bm, (short)0, Cm, false, false);
#else
        v8f Dm;
#pragma unroll
        for (int r = 0; r < 8; ++r) {
            int i = ti * 16 + r + ((lane >> 4) << 3);
            int j = tj * 16 + l15;
            Dm[r] = s_area[i] + s_area[j];
        }
#endif
#pragma unroll
        for (int r = 0; r < 8; ++r) {
            int i = ti * 16 + r + ((lane >> 4) << 3);   // C/D layout: lane>=16 -> M+8
            int j = tj * 16 + l15;
            if (i < TOP_K && j < TOP_K && j > i) {
                float xx1 = fmaxf(s_bx1[i], s_bx1[j]);
                float yy1 = fmaxf(s_by1[i], s_by1[j]);
                float xx2 = fminf(s_bx2[i], s_bx2[j]);
                float yy2 = fminf(s_by2[i], s_by2[j]);
                float inter = (xx2 - xx1) * (yy2 - yy1);   // NOT clamped (faithful)
                float uni   = Dm[r] - inter;               // area_i + area_j - inter
                float iou   = inter / uni;
                if (iou > NMS_THRESH)
                    atomicOr(&s_mask[i * MASKW + (j >> 5)], 1u << (j & 31));
            }
        }
    }
    __syncthreads();

    // ---- phase 5: sequential suppression scan (exact ref semantics) and
    //      stable compaction positions; one lane, bit-packed ----
    if (tid == 0) {
        unsigned supp[MASKW];
#pragma unroll
        for (int w = 0; w < MASKW; ++w) supp[w] = 0u;
        for (int i = 0; i < TOP_K; ++i)
            if (!(s_topv[i] > CONF_THRESH)) supp[i >> 5] |= 1u << (i & 31);
        int cnt = 0;
        for (int i = 0; i < TOP_K; ++i) {
            bool sup = (supp[i >> 5] >> (i & 31)) & 1u;
            if (!sup) {
                s_pos[i] = cnt++;
#pragma unroll
                for (int w = 0; w < MASKW; ++w) supp[w] |= s_mask[i * MASKW + w];
            } else {
                s_pos[i] = -1;
            }
        }
    }
    __syncthreads();

    // ---- phase 6: scatter kept rows (score, x1, y1, x2, y2); rest stay 0 ----
    if (tid < TOP_K) {
        int p = s_pos[tid];
        if (p >= 0) {
            size_t base = (((size_t)b * NCLS + (cc + 1)) * TOP_K + (size_t)p) * 5;
            out[base + 0] = s_topv[tid];
            out[base + 1] = s_bx1[tid];
            out[base + 2] = s_by1[tid];
            out[base + 3] = s_bx2[tid];
            out[base + 4] = s_by2[tid];
        }
    }
}

// ---------------------------------------------------------------------------
extern "C" void kernel_launch(void* const* d_in, const int* in_sizes, int n_in,
                              void* d_out, int out_size, void* d_ws, size_t ws_size,
                              hipStream_t stream) {
    const float* loc  = (const float*)d_in[0];   // [B, D, 4]
    const float* conf = (const float*)d_in[1];   // [B, D, 21]
    const float* dbox = (const float*)d_in[2];   // [D, 4]
    float* out = (float*)d_out;                  // [B, 21, 200, 5]

    const int D = in_sizes[2] / 4;
    const int B = in_sizes[0] / (4 * D);

    float* ws_boxes  = (float*)d_ws;                       // B*D*4 floats
    float* ws_scores = ws_boxes + (size_t)B * D * 4;       // B*20*D floats

    // zero the output (class 0 stays empty; non-kept rows stay 0)
    {
        int n = out_size;
        int blocks = (n + 255) / 256;
        if (blocks > 2048) blocks = 2048;
        ssd_zero_kernel<<<blocks, 256, 0, stream>>>(out, n);
    }

    // decode + softmax
    {
        int n = B * D;
        ssd_decode_softmax_kernel<<<(n + 255) / 256, 256, 0, stream>>>(
            loc, conf, dbox, ws_boxes, ws_scores, B, D);
    }

    // per-(batch,class) top-k + NMS + compaction
    ssd_topk_nms_kernel<<<B * NFG, 256, 0, stream>>>(ws_boxes, ws_scores, out, B, D);
}